// Polynomial_69114613728500
// MI455X (gfx1250) — compile-verified
//
#include <hip/hip_runtime.h>

// ---------------------------------------------------------------------------
// Polynomial feature expansion + GEMM for MI455X (gfx1250, wave32, WMMA).
//
// out[N,64] = basis(x)[N,164] @ W.T[164,64] + bias
//
// Strategy:
//  * fp32 WMMA (V_WMMA_F32_16X16X4_F32) -> exact fp32 GEMM, 1024 MAC/instr.
//  * basis built incrementally (deg1 -> deg2 -> deg3) = ~164 mults/row,
//    all register-resident via constexpr index tables + full unroll.
//  * reference term ordering handled by permuting the WEIGHTS once in a
//    prepack kernel, which also swizzles them into WMMA B-fragment layout.
// ---------------------------------------------------------------------------

typedef __attribute__((ext_vector_type(2))) float v2f;
typedef __attribute__((ext_vector_type(8))) float v8f;

#define IN_F        8
#define NUM_TERMS   164
#define NUM_CHUNKS  41          // 164 / 4 exactly
#define OUTF        64
#define NTILES      4           // 64 / 16
#define ROWS_PER_WAVE 32
#define WAVES       4
#define BLOCK_T     (WAVES * 32)
#define ROWS_PER_BLOCK (WAVES * ROWS_PER_WAVE)
#define BPACK_ELEMS (NUM_CHUNKS * NTILES * 32 * 2)   // 10496 floats = 41 KB

static_assert(NUM_CHUNKS * 4 == NUM_TERMS, "chunking must be exact");

// deg-3 term t (in OUR order) = d2[TRI.p[t]] * x[TRI.k[t]]
struct Tri { short p[120]; short k[120]; };
constexpr Tri makeTri() {
  Tri t{};
  int idx = 0;
  for (int i = 0; i < 8; ++i)
    for (int j = i; j < 8; ++j)
      for (int k = j; k < 8; ++k) {
        t.p[idx] = (short)(8 * i - i * (i - 1) / 2 + (j - i)); // pair index of (i,j)
        t.k[idx] = (short)k;
        ++idx;
      }
  return t;
}
constexpr Tri TRI = makeTri();

// ---------------------------------------------------------------------------
// Prepack: reproduce the reference meshgrid term enumeration, map each
// reference term to our degree-ordered index, and emit weights permuted +
// swizzled into WMMA B-fragment layout:
//   bpack[((chunk*4+ntile)*32 + lane)*2 + v] = W[n][ref_term(chunk*4+kloc)]
//   lane<16: n = ntile*16+lane,      kloc = v      (K0,K1)
//   lane>=16: n = ntile*16+lane-16,  kloc = 2+v    (K2,K3)
// Runs every launch (deterministic, no caching); ~0.1 Mops total.
// ---------------------------------------------------------------------------
__global__ void prepack_kernel(const float* __restrict__ weight,
                               float* __restrict__ bpack) {
  __shared__ unsigned refE[NUM_TERMS];
  __shared__ int cnt[256];
  __shared__ short our2ref[NUM_TERMS];
  const int t = threadIdx.x;

  // Reference flatten order (np.meshgrid 'xy' + .T + reshape):
  //   e1 = i&3, e0 = (i>>2)&3, e2..e7 = (i>>4)&3 ... (i>>14)&3
  // keep iff 1 <= sum <= 3  (exactly 164 terms).
  int c = 0;
  for (int i = t * 256; i < (t + 1) * 256; ++i) {
    int s = (i & 3) + ((i >> 2) & 3) + ((i >> 4) & 3) + ((i >> 6) & 3) +
            ((i >> 8) & 3) + ((i >> 10) & 3) + ((i >> 12) & 3) + ((i >> 14) & 3);
    if (s >= 1 && s <= 3) ++c;
  }
  cnt[t] = c;
  __syncthreads();
  if (t == 0) {
    int run = 0;
    for (int i = 0; i < 256; ++i) { int v = cnt[i]; cnt[i] = run; run += v; }
  }
  __syncthreads();
  int off = cnt[t];
  for (int i = t * 256; i < (t + 1) * 256; ++i) {
    int e[8];
    e[1] = i & 3;         e[0] = (i >> 2) & 3;  e[2] = (i >> 4) & 3;
    e[3] = (i >> 6) & 3;  e[4] = (i >> 8) & 3;  e[5] = (i >> 10) & 3;
    e[6] = (i >> 12) & 3; e[7] = (i >> 14) & 3;
    int s = e[0] + e[1] + e[2] + e[3] + e[4] + e[5] + e[6] + e[7];
    if (s >= 1 && s <= 3) {
      unsigned pk = 0;
      for (int f = 0; f < 8; ++f) pk |= (unsigned)e[f] << (4 * f);
      refE[off++] = pk;
    }
  }
  __syncthreads();

  // ref term -> our degree-ordered index (brute-force, tiny)
  if (t < NUM_TERMS) {
    unsigned pk = refE[t];
    int a[3]; int s = 0;
    for (int f = 0; f < 8; ++f) {
      int e = (pk >> (4 * f)) & 15;
      for (int r = 0; r < e; ++r) a[s++] = f;   // sorted multiset of factors
    }
    int ours = 0;
    if (s == 1) {
      ours = a[0];
    } else if (s == 2) {
      int idx = 0;
      for (int i = 0; i < 8; ++i)
        for (int j = i; j < 8; ++j) {
          if (i == a[0] && j == a[1]) ours = 8 + idx;
          ++idx;
        }
    } else {
      int idx = 0;
      for (int i = 0; i < 8; ++i)
        for (int j = i; j < 8; ++j)
          for (int k = j; k < 8; ++k) {
            if (i == a[0] && j == a[1] && k == a[2]) ours = 44 + idx;
            ++idx;
          }
    }
    our2ref[ours] = (short)t;
  }
  __syncthreads();

  // emit packed, swizzled B fragments
  for (int i = t; i < BPACK_ELEMS; i += 256) {
    int v  = i & 1;
    int l  = (i >> 1) & 31;
    int nt = (i >> 6) & 3;
    int ch = i >> 8;
    int n    = nt * 16 + (l & 15);
    int kloc = ((l >> 4) << 1) + v;
    int ourt = ch * 4 + kloc;
    int rt   = our2ref[ourt];
    bpack[i] = weight[n * NUM_TERMS + rt];
  }
}

// ---------------------------------------------------------------------------
// Main kernel: 4 waves / block, 32 rows / wave, full 64 outputs per wave.
// ---------------------------------------------------------------------------
__global__ __launch_bounds__(BLOCK_T)
void poly_wmma_kernel(const float* __restrict__ x,
                      const float* __restrict__ bpack,
                      const float* __restrict__ bias,
                      float* __restrict__ out, int nRows) {
  __shared__ float sB[BPACK_ELEMS];                  // 41 KB packed weights
  __shared__ float sA[WAVES][ROWS_PER_WAVE * 4];     // 512 B / wave A staging

  const int tid = threadIdx.x;
  {
    const float4* src = (const float4*)bpack;
    float4* dst = (float4*)sB;
    for (int i = tid; i < BPACK_ELEMS / 4; i += BLOCK_T) dst[i] = src[i];
  }
  __syncthreads();

  const int wave = tid >> 5;
  const int lane = tid & 31;
  const int rowBase = (blockIdx.x * WAVES + wave) * ROWS_PER_WAVE;
  if (rowBase >= nRows) return;    // uniform per wave -> EXEC stays full

  int myRow = rowBase + lane;
  if (myRow >= nRows) myRow = nRows - 1;             // clamp load, mask store

  const float4* xp = (const float4*)(x + (size_t)myRow * IN_F);
  float4 xa = xp[0];
  float4 xb = xp[1];
  float xv[8] = {xa.x, xa.y, xa.z, xa.w, xb.x, xb.y, xb.z, xb.w};

  // all 36 degree-2 monomials, register resident
  float d2[36];
  {
    int p = 0;
#pragma unroll
    for (int i = 0; i < 8; ++i)
#pragma unroll
      for (int j = i; j < 8; ++j) d2[p++] = xv[i] * xv[j];
  }

  v8f acc[8];                                        // [mtile*4 + ntile]
#pragma unroll
  for (int m = 0; m < 8; ++m) acc[m] = {};

  float* sAw = sA[wave];
  const int nc = lane & 15;
  const int half = lane >> 4;

  // one K-chunk: stage 4 terms/row to LDS, load A/B fragments, 8 WMMAs
  auto mma4 = [&](int c, float t0, float t1, float t2, float t3) {
    ((float4*)sAw)[lane] = make_float4(t0, t1, t2, t3);   // ds_store_b128
    // A frag: lanes 0-15 -> K0,K1 of row nc; lanes 16-31 -> K2,K3
    v2f a0 = *(const v2f*)(sAw + nc * 4 + half * 2);            // rows 0-15
    v2f a1 = *(const v2f*)(sAw + (16 + nc) * 4 + half * 2);     // rows 16-31
#pragma unroll
    for (int nt = 0; nt < NTILES; ++nt) {
      v2f b = *(const v2f*)(sB + ((c * NTILES + nt) * 32 + lane) * 2);
      acc[nt] = __builtin_amdgcn_wmma_f32_16x16x4_f32(
          false, a0, false, b, (short)0, acc[nt], false, false);
      acc[4 + nt] = __builtin_amdgcn_wmma_f32_16x16x4_f32(
          false, a1, false, b, (short)0, acc[4 + nt], false, false);
    }
  };

  // chunks 0-1: degree-1 terms (x itself)
  mma4(0, xv[0], xv[1], xv[2], xv[3]);
  mma4(1, xv[4], xv[5], xv[6], xv[7]);
  // chunks 2-10: degree-2 terms
#pragma unroll
  for (int c = 2; c < 11; ++c) {
    const int b = (c - 2) * 4;
    mma4(c, d2[b], d2[b + 1], d2[b + 2], d2[b + 3]);
  }
  // chunks 11-40: degree-3 terms = d2 * x (one mult each; indices constexpr)
#pragma unroll
  for (int c = 11; c < 41; ++c) {
    const int u = (c - 11) * 4;
    mma4(c,
         d2[TRI.p[u + 0]] * xv[TRI.k[u + 0]],
         d2[TRI.p[u + 1]] * xv[TRI.k[u + 1]],
         d2[TRI.p[u + 2]] * xv[TRI.k[u + 2]],
         d2[TRI.p[u + 3]] * xv[TRI.k[u + 3]]);
  }

  // epilogue: bias + store. C/D layout: lane<16 -> M=j, lane>=16 -> M=j+8.
  float bv[4];
#pragma unroll
  for (int nt = 0; nt < 4; ++nt) bv[nt] = bias[nt * 16 + nc];

#pragma unroll
  for (int mt = 0; mt < 2; ++mt)
#pragma unroll
    for (int nt = 0; nt < 4; ++nt)
#pragma unroll
      for (int j = 0; j < 8; ++j) {
        int row = rowBase + mt * 16 + half * 8 + j;
        if (row < nRows)
          out[(size_t)row * OUTF + nt * 16 + nc] = acc[mt * 4 + nt][j] + bv[nt];
      }
}

// ---------------------------------------------------------------------------
extern "C" void kernel_launch(void* const* d_in, const int* in_sizes, int n_in,
                              void* d_out, int out_size, void* d_ws, size_t ws_size,
                              hipStream_t stream) {
  const float* x      = (const float*)d_in[0];
  const float* weight = (const float*)d_in[1];
  const float* bias   = (const float*)d_in[2];
  float* out   = (float*)d_out;
  float* bpack = (float*)d_ws;          // 41984 bytes used

  const int nRows = in_sizes[0] / IN_F;

  prepack_kernel<<<1, 256, 0, stream>>>(weight, bpack);

  const int grid = (nRows + ROWS_PER_BLOCK - 1) / ROWS_PER_BLOCK;
  poly_wmma_kernel<<<grid, BLOCK_T, 0, stream>>>(x, bpack, bias, out, nRows);
}